// layer2EfficientKAN_58944131170347
// MI455X (gfx1250) — compile-verified
//
#include <hip/hip_runtime.h>
#include <hip/hip_bf16.h>
#include <math.h>

typedef __attribute__((ext_vector_type(2))) float v2f;
typedef __attribute__((ext_vector_type(8))) float v8f;

#define NB     8      // GRID_SIZE + SPLINE_ORDER = 5 + 3 basis functions
#define BLOCK  256
#define WAVES  (BLOCK / 32)

__device__ __forceinline__ float silu_f(float v) {
    // x * sigmoid(x). Denominator 1+exp(-v) is in (1, inf) -> hardware
    // v_rcp_f32 is safe; avoids the IEEE div_scale/div_fixup expansion.
    float d = 1.0f + __expf(-v);
    return v * __builtin_amdgcn_rcpf(d);
}

// Local-support cardinal cubic B-spline basis on the uniform grid
// knots g_k = -1 + 0.4*(k-3), k = 0..11 ; h = 0.4 ; support [-2.2, 2.2)
// For x in knot interval I (0..10), nonzero bases are j = I-3..I with
// values p3..p0; outside the knot span all bases are exactly 0 (matches
// the reference's indicator initialization of the Cox-de-Boor recursion).
__device__ __forceinline__ void basis4(float x, int& I,
                                       float& p0, float& p1, float& p2, float& p3) {
    float t  = (x + 2.2f) * 2.5f;                 // (x - g0) / h
    float fI = floorf(t);
    float m  = (t >= 0.0f && t < 11.0f) ? 1.0f : 0.0f;
    fI = fminf(fmaxf(fI, 0.0f), 10.0f);           // clamp for safe addressing
    I  = (int)fI;
    float u  = t - fI;
    float u2 = u * u, u3 = u2 * u;
    const float k6 = 1.0f / 6.0f;
    p0 = m * (k6 * u3);                                      // basis j = I
    p1 = m * (k6 * (-3.f*u3 + 3.f*u2 + 3.f*u + 1.f));        // basis j = I-1
    p2 = m * (k6 * ( 3.f*u3 - 6.f*u2 + 4.f));                // basis j = I-2
    float w = 1.f - u;
    p3 = m * (k6 * (w*w*w));                                 // basis j = I-3
}

__global__ __launch_bounds__(BLOCK)
void kan2_wmma_kernel(const float* __restrict__ x_in,
                      const float* __restrict__ bw1,   // (8,1)
                      const float* __restrict__ sw1,   // (8,1,8)
                      const float* __restrict__ sc1,   // (8,1)
                      const float* __restrict__ bw2,   // (1,8)
                      const float* __restrict__ sw2,   // (1,8,8)
                      const float* __restrict__ sc2,   // (1,8)
                      float* __restrict__ out, int n)
{
    // WMMA B-matrix: Bpair[p][n] = {B[2p][n], B[2p+1][n]}, B[k][n] = sw1[n][k]*sc1[n]
    __shared__ v2f  Bpair[4][16];
    // Layer-2 scaled weights, zero-padded: W2pad[f][m] = w2s[f][m-4] for m in [4,12)
    __shared__ float W2pad[8][16];
    // Per-wave scratch: padded basis rows (A-stage), then reused as D-tile
    __shared__ __align__(16) float tile[WAVES][32][16];

    const int tid  = threadIdx.x;
    const int wave = tid >> 5;
    const int lane = tid & 31;
    const int half = lane >> 4;      // 0: lanes 0-15, 1: lanes 16-31
    const int r0   = lane & 15;
    const int s    = blockIdx.x * BLOCK + tid;

    // ---- stage scaled / padded weights into LDS (once per block) ----
    if (tid < 64) {
        int p = tid >> 4, nn = tid & 15;
        v2f v; v.x = 0.f; v.y = 0.f;
        if (nn < NB) {
            float sc = sc1[nn];
            v.x = sw1[nn * NB + 2 * p]     * sc;
            v.y = sw1[nn * NB + 2 * p + 1] * sc;
        }
        Bpair[p][nn] = v;
    } else if (tid < 192) {
        int t2 = tid - 64; int f = t2 >> 4, mm = t2 & 15;
        float v = 0.f;
        if (mm >= 4 && mm < 12) v = sw2[f * NB + (mm - 4)] * sc2[f];
        W2pad[f][mm] = v;
    }
    __syncthreads();

    const float x = (s < n) ? x_in[s] : 0.0f;

    // ---- layer-1 basis (local support, 4 nonzero taps) ----
    int I1; float p0, p1, p2, p3;
    basis4(x, I1, p0, p1, p2, p3);

    // Dense padded basis row: row[m], B_j at m = j+4; nonzeros at m = I1+1..I1+4
    float* row = &tile[wave][lane][0];
    float4 z4 = make_float4(0.f, 0.f, 0.f, 0.f);
    reinterpret_cast<float4*>(row)[0] = z4;
    reinterpret_cast<float4*>(row)[1] = z4;
    reinterpret_cast<float4*>(row)[2] = z4;
    reinterpret_cast<float4*>(row)[3] = z4;
    row[I1 + 1] = p3;  // j = I-3
    row[I1 + 2] = p2;  // j = I-2
    row[I1 + 3] = p1;  // j = I-1
    row[I1 + 4] = p0;  // j = I
    __syncthreads();

    // ---- WMMA: h_spline = bases(16x8) x W1s(8x8), two K=4 steps per tile ----
    // A layout (f32 16x4): VGPR v holds K = v + (lane>=16 ? 2 : 0), row M = lane%16
    v2f a00 = *reinterpret_cast<v2f*>(&tile[wave][r0][4 + 2 * half]);       // tile0 K0..3
    v2f a01 = *reinterpret_cast<v2f*>(&tile[wave][r0][8 + 2 * half]);       // tile0 K4..7
    v2f a10 = *reinterpret_cast<v2f*>(&tile[wave][16 + r0][4 + 2 * half]);  // tile1 K0..3
    v2f a11 = *reinterpret_cast<v2f*>(&tile[wave][16 + r0][8 + 2 * half]);  // tile1 K4..7
    v2f b0 = Bpair[half][r0];         // B rows K0..3
    v2f b1 = Bpair[2 + half][r0];     // B rows K4..7

    v8f c0 = {0.f,0.f,0.f,0.f,0.f,0.f,0.f,0.f};
    v8f c1 = {0.f,0.f,0.f,0.f,0.f,0.f,0.f,0.f};
    c0 = __builtin_amdgcn_wmma_f32_16x16x4_f32(false, a00, false, b0, (short)0, c0, false, false);
    c0 = __builtin_amdgcn_wmma_f32_16x16x4_f32(false, a01, false, b1, (short)0, c0, false, false);
    c1 = __builtin_amdgcn_wmma_f32_16x16x4_f32(false, a10, false, b0, (short)0, c1, false, false);
    c1 = __builtin_amdgcn_wmma_f32_16x16x4_f32(false, a11, false, b1, (short)0, c1, false, false);

    __syncthreads();
    // D layout: VGPR r, lanes 0-15 -> (M=r, N=lane), lanes 16-31 -> (M=r+8, N=lane-16)
    #pragma unroll
    for (int r = 0; r < 8; ++r) {
        tile[wave][r + 8 * half][r0]      = c0[r];
        tile[wave][16 + r + 8 * half][r0] = c1[r];
    }
    __syncthreads();

    // ---- layer-1 base path + gather h, then layer 2 per lane ----
    const float sx = silu_f(x);
    float hv[8];
    #pragma unroll
    for (int o = 0; o < 8; ++o)
        hv[o] = tile[wave][lane][o] + sx * bw1[o];

    float res = 0.f;
    #pragma unroll
    for (int f = 0; f < 8; ++f) {
        float v = hv[f];
        res += bw2[f] * silu_f(v);
        int I2; float q0, q1, q2, q3;
        basis4(v, I2, q0, q1, q2, q3);
        const float* wr = &W2pad[f][0];
        // window [I2+1, I2+4] pairs with (p3, p2, p1, p0)
        res += q3 * wr[I2 + 1] + q2 * wr[I2 + 2] + q1 * wr[I2 + 3] + q0 * wr[I2 + 4];
    }

    if (s < n) out[s] = res;
}

extern "C" void kernel_launch(void* const* d_in, const int* in_sizes, int n_in,
                              void* d_out, int out_size, void* d_ws, size_t ws_size,
                              hipStream_t stream) {
    const float* x   = (const float*)d_in[0];
    const float* bw1 = (const float*)d_in[1];
    const float* sw1 = (const float*)d_in[2];
    const float* sc1 = (const float*)d_in[3];
    const float* bw2 = (const float*)d_in[4];
    const float* sw2 = (const float*)d_in[5];
    const float* sc2 = (const float*)d_in[6];
    const int n = in_sizes[0];               // B (x is (B,1))
    const int blocks = (n + BLOCK - 1) / BLOCK;
    kan2_wmma_kernel<<<blocks, BLOCK, 0, stream>>>(x, bw1, sw1, sc1, bw2, sw2, sc2,
                                                   (float*)d_out, n);
}